// NounPhraseModel_81252191306701
// MI455X (gfx1250) — compile-verified
//
#include <hip/hip_runtime.h>
#include <hip/hip_bf16.h>

// NounPhraseModel.attention on gfx1250 (MI455X), fp32 WMMA path.
//
// Pipeline:
//   k0: group start offsets from sorted gid (binary search per group)
//   k1: qp = q @ Wq + bq        (4096 x 1024 x 512)   v_wmma_f32_16x16x4_f32
//   k2: kp = k @ Wk + bk        (16384 x 512 x 512)   v_wmma_f32_16x16x4_f32
//   k3: per-group attention: S = qp Kp^T/sqrt(d) - (exp(50(1-m))-1),
//       row softmax (LDS staged), out = P V                   (WMMA both GEMMs)

typedef __attribute__((ext_vector_type(2))) float v2f;
typedef __attribute__((ext_vector_type(8))) float v8f;

#define NQTOT 4096
#define NGRP  256
#define LG    64
#define DH    512

static __device__ __forceinline__ v8f wmma_f32(v2f a, v2f b, v8f c) {
  // D = A(16x4,f32) * B(4x16,f32) + C(16x16,f32)
  return __builtin_amdgcn_wmma_f32_16x16x4_f32(
      /*neg_a=*/false, a, /*neg_b=*/false, b,
      /*c_mod=*/(short)0, c, /*reuse_a=*/false, /*reuse_b=*/false);
}

// ---------------------------------------------------------------------------
// k0: gstart[g] = lower_bound(gid, g); gstart[NGRP] = NQTOT
// ---------------------------------------------------------------------------
__global__ void group_starts_kernel(const int* __restrict__ gid,
                                    int* __restrict__ gstart) {
  const int g = threadIdx.x;  // 0..255
  int lo = 0, hi = NQTOT;
  while (lo < hi) {
    int mid = (lo + hi) >> 1;
    if (gid[mid] < g) lo = mid + 1; else hi = mid;
  }
  gstart[g] = lo;
  if (g == 0) gstart[NGRP] = NQTOT;
}

// ---------------------------------------------------------------------------
// k1/k2: C[M,N] = A[M,K] @ B[K,N] + bias[N], fp32 WMMA 16x16x4.
// Block = 8 waves; wave w owns output tile (blockIdx.x*16, (blockIdx.y*8+w)*16).
// A-frag layout (ISA 7.12.2, 32-bit A 16x4): lane[0..15] -> M=lane, VGPR0=K0,
// VGPR1=K1; lane[16..31] -> K2,K3.  B-frag is the dual with N across lanes.
// ---------------------------------------------------------------------------
__global__ __launch_bounds__(256) void gemm_bias_wmma(
    const float* __restrict__ A, const float* __restrict__ B,
    const float* __restrict__ bias, float* __restrict__ C,
    int M, int N, int K) {
  const int lane = threadIdx.x & 31;
  const int wave = threadIdx.x >> 5;
  const int l15  = lane & 15;
  const int kh   = lane >> 4;                 // K half-pair select (0 or 1)
  const int m0   = blockIdx.x * 16;
  const int n0   = (blockIdx.y * 8 + wave) * 16;

  const float* ap = A + (long)(m0 + l15) * K + kh * 2;
  const float* bp = B + (long)(kh * 2) * N + (n0 + l15);

  v8f acc = {};
#pragma unroll 8
  for (int k = 0; k < K; k += 4) {
    v2f a = *(const v2f*)(ap + k);            // A[m][k+2kh], A[m][k+2kh+1]
    v2f b;
    b.x = bp[(long)k * N];                    // B[k+2kh][n]
    b.y = bp[(long)(k + 1) * N];              // B[k+2kh+1][n]
    acc = wmma_f32(a, b, acc);
  }

  const float bv = bias[n0 + l15];
#pragma unroll
  for (int r = 0; r < 8; ++r) {               // C row = m0 + r + 8*kh
    C[(long)(m0 + r + kh * 8) * N + n0 + l15] = acc[r] + bv;
  }
}

// ---------------------------------------------------------------------------
// k3: one block (8 waves) per group. Each wave handles one 16-query tile:
//   S(16x64) = Qtile(16x512) Kp_g^T / sqrt(512) - (exp(50(1-m))-1)
//   P = softmax_rows(S)  (staged in per-wave LDS tile)
//   Out(16x512) = P(16x64) V_g(64x512)
// ---------------------------------------------------------------------------
__global__ __launch_bounds__(256) void attn_kernel(
    const float* __restrict__ qp, const float* __restrict__ kp,
    const float* __restrict__ v,  const float* __restrict__ m,
    const int* __restrict__ gstart, float* __restrict__ out) {
  __shared__ float sS[8 * 16 * LG];           // 32 KB: per-wave 16x64 prob tile

  const int g    = blockIdx.x;
  const int lane = threadIdx.x & 31;
  const int wave = threadIdx.x >> 5;
  const int l15  = lane & 15;
  const int kh   = lane >> 4;

  const int start  = gstart[g];
  const int end    = gstart[g + 1];
  const int ntiles = (end - start + 15) >> 4;
  if (ntiles == 0) return;

  float* S = sS + wave * (16 * LG);
  const float scale = 0.04419417382415922f;   // 1/sqrt(512)
  const float* kg = kp + (long)g * LG * DH;
  const float* vg = v  + (long)g * LG * DH;

  // Exponential mask bias for this lane's score columns (4 N-tiles of 16).
  float mbias[4];
#pragma unroll
  for (int j = 0; j < 4; ++j) {
    float mm = m[g * LG + j * 16 + l15];
    mbias[j] = __expf(50.0f * (1.0f - mm)) - 1.0f;
  }

  for (int t = wave; t < ntiles; t += 8) {
    const int qbase = start + t * 16;

    // ---- scores: 4 WMMA column tiles over K = DH ----
    int arow = qbase + l15;
    if (arow > NQTOT - 1) arow = NQTOT - 1;   // tail rows: computed, not stored
    const float* aq = qp + (long)arow * DH + kh * 2;

    v8f acc[4];
#pragma unroll
    for (int j = 0; j < 4; ++j) acc[j] = (v8f){};

#pragma unroll 4
    for (int d = 0; d < DH; d += 4) {
      v2f a = *(const v2f*)(aq + d);
#pragma unroll
      for (int j = 0; j < 4; ++j) {
        // B[dd][l] = kp[g][l][dd] -> contiguous K-pair per lane
        v2f b = *(const v2f*)(kg + (long)(j * 16 + l15) * DH + d + kh * 2);
        acc[j] = wmma_f32(a, b, acc[j]);
      }
    }

    // scale + mask bias -> LDS (C layout: VGPR r holds row r + 8*kh)
#pragma unroll
    for (int j = 0; j < 4; ++j) {
#pragma unroll
      for (int r = 0; r < 8; ++r) {
        S[(r + kh * 8) * LG + j * 16 + l15] = acc[j][r] * scale - mbias[j];
      }
    }

    // ---- row softmax over L=64 (lanes 0..15 each own one row) ----
    if (lane < 16) {
      float* row = S + lane * LG;
      float mx = -3.402823466e38f;
#pragma unroll 8
      for (int l = 0; l < LG; ++l) mx = fmaxf(mx, row[l]);
      float sum = 0.0f;
#pragma unroll 8
      for (int l = 0; l < LG; ++l) {
        float e = __expf(row[l] - mx);
        row[l] = e;
        sum += e;
      }
      float inv = 1.0f / sum;
#pragma unroll 8
      for (int l = 0; l < LG; ++l) row[l] *= inv;
    }

    // ---- out tile: P(16x64) @ V_g(64x512), 32 WMMA N-tiles ----
    for (int jn = 0; jn < DH / 16; ++jn) {
      const int n0 = jn * 16;
      v8f oc = {};
#pragma unroll
      for (int kk = 0; kk < LG; kk += 4) {
        v2f a = *(const v2f*)(S + l15 * LG + kk + kh * 2);  // P[m][k pair]
        v2f b;
        b.x = vg[(long)(kk + kh * 2) * DH + n0 + l15];      // V[k][n]
        b.y = vg[(long)(kk + kh * 2 + 1) * DH + n0 + l15];
        oc = wmma_f32(a, b, oc);
      }
#pragma unroll
      for (int r = 0; r < 8; ++r) {
        const int rowi = qbase + r + kh * 8;
        if (rowi < end) out[(long)rowi * DH + n0 + l15] = oc[r];
      }
    }
  }
}

// ---------------------------------------------------------------------------
// Launch. Inputs: q, k, v, m, gid, Wq, bq, Wk, bk. Output: (4096, 512) f32.
// Workspace layout: [gstart 257 ints | qp 4096x512 f32 | kp 16384x512 f32]
// ---------------------------------------------------------------------------
extern "C" void kernel_launch(void* const* d_in, const int* in_sizes, int n_in,
                              void* d_out, int out_size, void* d_ws, size_t ws_size,
                              hipStream_t stream) {
  const float* q   = (const float*)d_in[0];
  const float* k   = (const float*)d_in[1];
  const float* v   = (const float*)d_in[2];
  const float* m   = (const float*)d_in[3];
  const int*   gid = (const int*)d_in[4];
  const float* Wq  = (const float*)d_in[5];
  const float* bq  = (const float*)d_in[6];
  const float* Wk  = (const float*)d_in[7];
  const float* bk  = (const float*)d_in[8];
  float* out = (float*)d_out;

  char* ws = (char*)d_ws;
  int*   gstart = (int*)ws;
  float* qp = (float*)(ws + 4096);
  float* kp = (float*)(ws + 4096 + (size_t)NQTOT * DH * sizeof(float));

  group_starts_kernel<<<1, 256, 0, stream>>>(gid, gstart);

  // qp = q @ Wq + bq : M=4096, N=512, K=1024
  gemm_bias_wmma<<<dim3(NQTOT / 16, DH / 128), 256, 0, stream>>>(
      q, Wq, bq, qp, NQTOT, DH, 2 * DH);

  // kp = k @ Wk + bk : M=16384, N=512, K=512
  gemm_bias_wmma<<<dim3((NGRP * LG) / 16, DH / 128), 256, 0, stream>>>(
      k, Wk, bk, kp, NGRP * LG, DH, DH);

  attn_kernel<<<NGRP, 256, 0, stream>>>(qp, kp, v, m, gstart, out);
}